// FrameReducer_515396075632
// MI455X (gfx1250) — compile-verified
//
#include <hip/hip_runtime.h>
#include <hip/hip_bf16.h>
#include <stdint.h>

// Problem constants from the reference: N, T, C, V = 32, 2048, 512, 500
#define FR_N 32
#define FR_T 2048
#define FR_C 512
#define FR_V 500
#define LOG_0_9 (-0.10536051565782628f)

typedef float vfloat4 __attribute__((ext_vector_type(4)));

// ---------------------------------------------------------------------------
// Phase 1: per-batch-row mask + stable compaction map via block scan.
// One 256-thread block per n; each thread owns 8 consecutive t values.
// ---------------------------------------------------------------------------
__global__ void fr_mask_scan_kernel(const float* __restrict__ ctc,
                                    const int* __restrict__ x_lens,
                                    const int* __restrict__ blank_id_p,
                                    int* __restrict__ src_idx,
                                    int* __restrict__ out_len_i,
                                    float* __restrict__ out_len_f) {
    const int n    = blockIdx.x;
    const int tid  = threadIdx.x;           // 0..255
    const int PER  = FR_T / 256;            // 8
    const int blank = blank_id_p[0];
    const int len   = x_lens[n];

    const float* row = ctc + (size_t)n * FR_T * FR_V;

    int keep[8];
    int cnt = 0;
#pragma unroll
    for (int k = 0; k < PER; ++k) {
        const int t = tid * PER + k;
        const bool m = (t < len) && (row[(size_t)t * FR_V + blank] < LOG_0_9);
        keep[k] = m ? 1 : 0;
        cnt += keep[k];
    }

    // Block-wide inclusive scan (Hillis-Steele, double barrier).
    __shared__ int s[256];
    s[tid] = cnt;
    __syncthreads();
#pragma unroll
    for (int off = 1; off < 256; off <<= 1) {
        const int v = (tid >= off) ? s[tid - off] : 0;
        __syncthreads();
        s[tid] += v;
        __syncthreads();
    }
    const int incl  = s[tid];
    const int total = s[255];
    int pos = incl - cnt;                   // exclusive prefix

    int* dst = src_idx + n * FR_T;
#pragma unroll
    for (int k = 0; k < PER; ++k) {
        if (keep[k]) {
            dst[pos++] = tid * PER + k;     // stable: ascending t within thread,
        }                                   // threads ordered by scan
    }

    if (tid == 0) {
        out_len_i[n] = total;
        out_len_f[n] = (float)total;        // tuple's 2nd output, flattened as float
    }
}

// ---------------------------------------------------------------------------
// Phase 2: row compaction. One 128-thread block per output row (n, j).
// Kept rows are DMA'd global -> LDS -> global with the gfx1250 async pipe
// (ASYNCcnt-tracked), using NT cache policy both ways: the 256 MiB stream
// has zero reuse, so keeping it out of the 192 MB L2 is strictly better.
// Tail rows are zero-filled with non-temporal 128-bit stores.
// The branch is block-uniform, so EXEC is all-ones for the async ops.
// ---------------------------------------------------------------------------
__global__ void fr_compact_kernel(const float* __restrict__ x,
                                  const int* __restrict__ src_idx,
                                  const int* __restrict__ out_len_i,
                                  float* __restrict__ out) {
    __shared__ char smem[128 * 16];         // 16 B private slot per thread

    const int row = blockIdx.x;             // 0 .. N*T-1
    const int n   = row >> 11;              // row / T
    const int j   = row & (FR_T - 1);       // row % T
    const int tid = threadIdx.x;            // 0..127; 128 * 16 B = 2 KB = one row

    float* drow = out + (size_t)row * FR_C;

    if (j < out_len_i[n]) {
        const int t = src_idx[n * FR_T + j];
        const float* srow = x + ((size_t)n * FR_T + (size_t)t) * FR_C;

        // Low 32 bits of the flat shared-aperture address == LDS byte offset
        // (CDNA5 ISA 10.2).
        uint32_t lds = (uint32_t)(size_t)(&smem[tid * 16]);
        uint64_t src = (uint64_t)(size_t)(srow + tid * 4);
        uint64_t dst = (uint64_t)(size_t)(drow + tid * 4);

        asm volatile(
            "global_load_async_to_lds_b128 %0, %1, off th:TH_LOAD_NT\n\t"
            "s_wait_asynccnt 0\n\t"
            "global_store_async_from_lds_b128 %2, %0, off th:TH_STORE_NT"
            :
            : "v"(lds), "v"(src), "v"(dst)
            : "memory");
        // s_endpgm performs an implicit wait-idle; LDS slot is never reused,
        // so no trailing s_wait_asynccnt is needed.
    } else {
        vfloat4 z = (vfloat4)0.0f;
        __builtin_nontemporal_store(z, (vfloat4*)(drow + tid * 4));
    }
}

// ---------------------------------------------------------------------------
// Harness entry point.
//   d_in[0]: x          [N*T*C] float32
//   d_in[1]: x_lens     [N]     int
//   d_in[2]: ctc_output [N*T*V] float32
//   d_in[3]: blank_id   [1]     int
//   d_out  : out [N*T*C] floats ++ out_lens [N] floats
//   d_ws   : src_idx [N*T] int, out_len [N] int  (~256 KiB)
// ---------------------------------------------------------------------------
extern "C" void kernel_launch(void* const* d_in, const int* in_sizes, int n_in,
                              void* d_out, int out_size, void* d_ws, size_t ws_size,
                              hipStream_t stream) {
    (void)in_sizes; (void)n_in; (void)out_size; (void)ws_size;

    const float* x       = (const float*)d_in[0];
    const int*   x_lens  = (const int*)d_in[1];
    const float* ctc     = (const float*)d_in[2];
    const int*   blank   = (const int*)d_in[3];

    float* out       = (float*)d_out;
    float* out_len_f = out + (size_t)FR_N * FR_T * FR_C;

    int* src_idx   = (int*)d_ws;
    int* out_len_i = src_idx + FR_N * FR_T;

    fr_mask_scan_kernel<<<FR_N, 256, 0, stream>>>(ctc, x_lens, blank,
                                                  src_idx, out_len_i, out_len_f);
    fr_compact_kernel<<<FR_N * FR_T, 128, 0, stream>>>(x, src_idx, out_len_i, out);
}